// CurvatureConstraint_44753559224572
// MI455X (gfx1250) — compile-verified
//
#include <hip/hip_runtime.h>

// ---------------------------------------------------------------------------
// CurvatureConstraint for MI455X (gfx1250, wave32).
// Gather/VALU-bound f32 reduction; WMMA F32_16X16X4 used as the deterministic
// reduction engine (D += ones(16x4) x partials(4x16)).
// ---------------------------------------------------------------------------

#define Wd 40
#define Hd 40
#define Dd 40
#define NCELL (Wd * Hd * Dd)   // 64000
#define NCFG  96
#define TCOL  256

// d_ws layout in 4-byte units:
//   [0,   384)  edges packed per (cfg,tri): e0 | e1<<4 | e2<<8      (int)
//   [384, 480)  meta per cfg: ntri | topo_col<<8                    (int)
//   [480, 736)  block partial sums, 256 floats (250 used, rest 0)   (float)
#define WS_EDGES 0
#define WS_META  384
#define WS_PART  480

typedef __attribute__((ext_vector_type(2))) float v2f;
typedef __attribute__((ext_vector_type(8))) float v8f;

// ---------------------------------------------------------------------------
// Init: reproduce numpy RandomState(0) tables on-device (legacy MT19937).
// Draw order in the reference: randint(0,256,96) -> rand(96,4,12) -> randint(1,5,96).
// ---------------------------------------------------------------------------
__global__ void curv_init_tables(int* __restrict__ ws)
{
    // zero the partial-sum region (entries 250..255 stay 0 for padding)
    ((float*)ws)[WS_PART + threadIdx.x] = 0.0f;
    if (threadIdx.x != 0) return;

    unsigned mt[624];
    int idx = 624;
    mt[0] = 0u;                                    // seed = 0
    for (int i = 1; i < 624; ++i)
        mt[i] = 1812433253u * (mt[i - 1] ^ (mt[i - 1] >> 30)) + (unsigned)i;

    auto next32 = [&]() -> unsigned {
        if (idx >= 624) {
            for (int i = 0; i < 624; ++i) {
                unsigned y = (mt[i] & 0x80000000u) | (mt[(i + 1) % 624] & 0x7fffffffu);
                unsigned v = mt[(i + 397) % 624] ^ (y >> 1);
                if (y & 1u) v ^= 2567483615u;
                mt[i] = v;
            }
            idx = 0;
        }
        unsigned y = mt[idx++];
        y ^= y >> 11;
        y ^= (y << 7)  & 2636928640u;
        y ^= (y << 15) & 4022730752u;
        y ^= y >> 18;
        return y;
    };
    auto nextd = [&]() -> double {
        unsigned a = next32() >> 5, b = next32() >> 6;
        return ((double)a * 67108864.0 + (double)b) / 9007199254740992.0;
    };

    // TOPO2TRI = randint(0, 256, 96): rng=255 (pow2 mask, no rejection)
    int topocol[NCFG];
    for (int c = 0; c < NCFG; ++c) topocol[c] = (int)(next32() & 255u);

    // TRI_EDGES = rand(96,4,12).argsort(-1)[..., :3] : 3 smallest, ascending
    for (int c = 0; c < NCFG; ++c) {
        for (int t = 0; t < 4; ++t) {
            double v[12];
            for (int e = 0; e < 12; ++e) v[e] = nextd();
            bool used[12] = {};
            int pick[3];
            for (int s = 0; s < 3; ++s) {
                int best = 0; double bv = 1e300;
                for (int e = 0; e < 12; ++e)
                    if (!used[e] && v[e] < bv) { bv = v[e]; best = e; }
                used[best] = true;
                pick[s] = best;
            }
            ws[WS_EDGES + c * 4 + t] = pick[0] | (pick[1] << 4) | (pick[2] << 8);
        }
    }

    // _NTRI = randint(1, 5, 96): rng=3 (pow2 mask, no rejection)
    for (int c = 0; c < NCFG; ++c) {
        int ntri = 1 + (int)(next32() & 3u);
        ws[WS_META + c] = ntri | (topocol[c] << 8);
    }
}

// ---------------------------------------------------------------------------
// Main kernel: one thread per cell, 256 threads/block, 250 blocks.
// ---------------------------------------------------------------------------
__global__ __launch_bounds__(256)
void curv_main(const float* __restrict__ off, const float* __restrict__ topo,
               const int* __restrict__ tabs, float* __restrict__ partials)
{
    __shared__ float4 sV[256 * 12];   // per-thread edge vertices (dynamic index -> LDS)
    __shared__ float  sPart[256];

    const int tid = threadIdx.x;
    const int n   = blockIdx.x * 256 + tid;
    const int i   = n / (Hd * Dd);
    const int rem = n - i * (Hd * Dd);
    const int j   = rem / Dd;
    const int k   = rem - j * Dd;

    // EDGES = [(dx,dy,dz,ax)] from the reference, compile-time constants
    constexpr int EDX[12] = {0,0,0,0, 0,1,0,1, 0,1,0,1};
    constexpr int EDY[12] = {0,1,0,1, 0,0,0,0, 0,0,1,1};
    constexpr int EDZ[12] = {0,0,1,1, 0,0,1,1, 0,0,0,0};
    constexpr int EAX[12] = {0,0,0,0, 1,1,1,1, 2,2,2,2};

    const int s3 = 41 * 41 * 41, s2 = 41 * 41, s1 = 41;
    float4* vrow = &sV[tid * 12];
    #pragma unroll
    for (int e = 0; e < 12; ++e) {
        const float o = off[EAX[e] * s3 + (i + EDX[e]) * s2 + (j + EDY[e]) * s1 + (k + EDZ[e])];
        float4 v;
        v.x = (float)EDX[e] + (EAX[e] == 0 ? o : 0.0f);
        v.y = (float)EDY[e] + (EAX[e] == 1 ? o : 0.0f);
        v.z = (float)EDZ[e] + (EAX[e] == 2 ? o : 0.0f);
        v.w = 0.0f;
        vrow[e] = v;           // each thread reads only its own row: no barrier needed
    }

    const int*   edgesT = tabs + WS_EDGES;
    const int*   metaT  = tabs + WS_META;
    const float* trow   = topo + (size_t)n * TCOL;

    float acc = 0.0f;
    for (int c = 0; c < NCFG; ++c) {
        const int meta = metaT[c];              // uniform -> scalar load
        const int ntri = meta & 0xff;
        float curv = 0.0f;
        float nx = 0.0f, ny = 0.0f, nz = 0.0f;
        #pragma unroll
        for (int t = 0; t < 4; ++t) {
            const int pk = edgesT[c * 4 + t];   // uniform -> scalar load
            const float4 a  = vrow[pk & 15];
            const float4 b  = vrow[(pk >> 4) & 15];
            const float4 cc = vrow[(pk >> 8) & 15];
            const float d1x = b.x - a.x,  d1y = b.y - a.y,  d1z = b.z - a.z;
            const float d2x = cc.x - a.x, d2y = cc.y - a.y, d2z = cc.z - a.z;
            float cx = d1y * d2z - d1z * d2y;
            float cy = d1z * d2x - d1x * d2z;
            float cz = d1x * d2y - d1y * d2x;
            const float inv = 1.0f / (sqrtf(cx * cx + cy * cy + cz * cz) + 1e-8f);
            cx *= inv; cy *= inv; cz *= inv;
            if (t > 0) {
                const float cs = nx * cx + ny * cy + nz * cz;
                curv += (t < ntri) ? (1.0f - cs) : 0.0f;   // PAIR_MASK
            }
            nx = cx; ny = cy; nz = cz;
        }
        acc += trow[meta >> 8] * curv;          // topo[n, TOPO2TRI[c]] (L2-resident gather)
    }

    sPart[tid] = acc;
    __syncthreads();

    // Wave 0 (full EXEC) reduces 256 partials: D += ones(16x4) x P(4x16), 4 chunks.
    if (tid < 32) {
        const v2f aone = {1.0f, 1.0f};
        v8f d = {0.0f, 0.0f, 0.0f, 0.0f, 0.0f, 0.0f, 0.0f, 0.0f};
        const int col = tid & 15;
        const int kr  = (tid < 16) ? 0 : 2;
        #pragma unroll
        for (int ch = 0; ch < 4; ++ch) {
            v2f bfrag;
            bfrag.x = sPart[ch * 64 + kr * 16 + col];
            bfrag.y = sPart[ch * 64 + (kr + 1) * 16 + col];
            d = __builtin_amdgcn_wmma_f32_16x16x4_f32(false, aone, false, bfrag,
                                                      (short)0, d, false, false);
        }
        // D row 0: lanes 0..15 hold per-column sums; sum them within the 16-group.
        float v = d[0];
        #pragma unroll
        for (int o2 = 8; o2 >= 1; o2 >>= 1) v += __shfl_xor(v, o2, 32);
        if (tid == 0) partials[blockIdx.x] = v;
    }
}

// ---------------------------------------------------------------------------
// Final: one wave reduces the 256 (250 live + 6 zero) block partials via WMMA.
// ---------------------------------------------------------------------------
__global__ void curv_final(const float* __restrict__ partials, float* __restrict__ out)
{
    const int tid = threadIdx.x;
    const v2f aone = {1.0f, 1.0f};
    v8f d = {0.0f, 0.0f, 0.0f, 0.0f, 0.0f, 0.0f, 0.0f, 0.0f};
    const int col = tid & 15;
    const int kr  = (tid < 16) ? 0 : 2;
    #pragma unroll
    for (int ch = 0; ch < 4; ++ch) {
        v2f bfrag;
        bfrag.x = partials[ch * 64 + kr * 16 + col];
        bfrag.y = partials[ch * 64 + (kr + 1) * 16 + col];
        d = __builtin_amdgcn_wmma_f32_16x16x4_f32(false, aone, false, bfrag,
                                                  (short)0, d, false, false);
    }
    float v = d[0];
    #pragma unroll
    for (int o2 = 8; o2 >= 1; o2 >>= 1) v += __shfl_xor(v, o2, 32);
    if (tid == 0) out[0] = v;
}

// ---------------------------------------------------------------------------
extern "C" void kernel_launch(void* const* d_in, const int* in_sizes, int n_in,
                              void* d_out, int out_size, void* d_ws, size_t ws_size,
                              hipStream_t stream)
{
    const float* off  = (const float*)d_in[0];   // [3, 41, 41, 41] f32
    const float* topo = (const float*)d_in[1];   // [64000, 256] f32
    int*   ws       = (int*)d_ws;
    float* partials = (float*)d_ws + WS_PART;

    curv_init_tables<<<1, 256, 0, stream>>>(ws);
    curv_main<<<NCELL / 256, 256, 0, stream>>>(off, topo, ws, partials);
    curv_final<<<1, 32, 0, stream>>>(partials, (float*)d_out);

    (void)in_sizes; (void)n_in; (void)out_size; (void)ws_size;
}